// KinematicsNetwork_40956808135374
// MI455X (gfx1250) — compile-verified
//
#include <hip/hip_runtime.h>
#include <math.h>

#define SE3_EPS 1e-5f

// ---------------------------------------------------------------------------
// Generic closed-form SE(3) exponential: twist[6]=(v,w) -> M[12] row-major [R|t]
// (used for the per-sample target pose and the batch-constant T0)
// ---------------------------------------------------------------------------
__device__ __forceinline__ void se3_exp6(const float tw[6], float M[12]) {
    const float vx = tw[0], vy = tw[1], vz = tw[2];
    const float wx = tw[3], wy = tw[4], wz = tw[5];
    const float th2 = wx * wx + wy * wy + wz * wz;
    const float th  = sqrtf(th2);
    float Ac, Bc, Cc;
    if (th < SE3_EPS) {
        Ac = 1.0f - th2 * (1.0f / 6.0f);
        Bc = 0.5f - th2 * (1.0f / 24.0f);
        Cc = (1.0f / 6.0f) - th2 * (1.0f / 120.0f);
    } else {
        float s, c;
        sincosf(th, &s, &c);
        const float r2 = 1.0f / th2;
        Ac = s / th;
        Bc = (1.0f - c) * r2;
        Cc = (th - s) * r2 / th;
    }
    const float xx = wx * wx, yy = wy * wy, zz = wz * wz;
    const float xy = wx * wy, xz = wx * wz, yz = wy * wz;
    M[0]  = 1.0f + Bc * (xx - th2);
    M[1]  = Bc * xy - Ac * wz;
    M[2]  = Bc * xz + Ac * wy;
    M[4]  = Bc * xy + Ac * wz;
    M[5]  = 1.0f + Bc * (yy - th2);
    M[6]  = Bc * yz - Ac * wx;
    M[8]  = Bc * xz - Ac * wy;
    M[9]  = Bc * yz + Ac * wx;
    M[10] = 1.0f + Bc * (zz - th2);
    const float V00 = 1.0f + Cc * (xx - th2);
    const float V01 = Cc * xy - Bc * wz;
    const float V02 = Cc * xz + Bc * wy;
    const float V10 = Cc * xy + Bc * wz;
    const float V11 = 1.0f + Cc * (yy - th2);
    const float V12 = Cc * yz - Bc * wx;
    const float V20 = Cc * xz - Bc * wy;
    const float V21 = Cc * yz + Bc * wx;
    const float V22 = 1.0f + Cc * (zz - th2);
    M[3]  = V00 * vx + V01 * vy + V02 * vz;
    M[7]  = V10 * vx + V11 * vy + V12 * vz;
    M[11] = V20 * vx + V21 * vy + V22 * vz;
}

// ---------------------------------------------------------------------------
// Per-joint constants, packed into five 16B-aligned float4 chunks (80B stride)
// so per-thread LDS broadcast reads are ds_load_b128.
//   u_k     = (u.x, u.y, u.z, k=|u|)
//   pd_poxy = (u_x^2-k^2, u_y^2-k^2, u_z^2-k^2, u_x u_y)
//   po_q1   = (u_x u_z, u_y u_z, (u x s_v).x, (u x s_v).y)
//   q1z_q2  = ((u x s_v).z, q2.x, q2.y, q2.z)  with q2 = u(u.s_v) - k^2 s_v
//   sv      = (s_v.x, s_v.y, s_v.z, 0)
// ---------------------------------------------------------------------------
struct JointC {
    float4 u_k;
    float4 pd_poxy;
    float4 po_q1;
    float4 q1z_q2;
    float4 sv;
};

// R = I + (A th)[u]x + (B th^2)(u u^T - k^2 I)
// t = th s_v + (B th^2)(u x s_v) + (C th^3)(u(u.s_v) - k^2 s_v)
__device__ __forceinline__ void joint_exp(const JointC* __restrict__ Jc,
                                          float thj, float M[12]) {
    const float4 uk  = Jc->u_k;
    const float4 pdp = Jc->pd_poxy;
    const float4 pq  = Jc->po_q1;
    const float4 qq  = Jc->q1z_q2;
    const float4 sv  = Jc->sv;

    const float th  = uk.w * fabsf(thj);   // |w| = k*|theta|
    const float th2 = th * th;
    float Ac, Bc, Cc;
    if (th < SE3_EPS) {
        Ac = 1.0f - th2 * (1.0f / 6.0f);
        Bc = 0.5f - th2 * (1.0f / 24.0f);
        Cc = (1.0f / 6.0f) - th2 * (1.0f / 120.0f);
    } else {
        float s, c;
        sincosf(th, &s, &c);
        const float r2 = 1.0f / th2;
        Ac = s / th;
        Bc = (1.0f - c) * r2;
        Cc = (th - s) * r2 / th;
    }
    const float q  = thj * thj;     // theta^2
    const float a  = Ac * thj;      // A * theta   (signed)
    const float b  = Bc * q;        // B * theta^2
    const float ct = Cc * q * thj;  // C * theta^3 (signed)
    M[0]  = 1.0f + b * pdp.x;
    M[1]  = b * pdp.w - a * uk.z;
    M[2]  = b * pq.x  + a * uk.y;
    M[4]  = b * pdp.w + a * uk.z;
    M[5]  = 1.0f + b * pdp.y;
    M[6]  = b * pq.y  - a * uk.x;
    M[8]  = b * pq.x  - a * uk.y;
    M[9]  = b * pq.y  + a * uk.x;
    M[10] = 1.0f + b * pdp.z;
    M[3]  = thj * sv.x + b * pq.z + ct * qq.y;
    M[7]  = thj * sv.y + b * pq.w + ct * qq.z;
    M[11] = thj * sv.z + b * qq.x + ct * qq.w;
}

// c = a o b  (3x4 rigid transforms, implicit bottom row [0 0 0 1]); c may alias a.
__device__ __forceinline__ void composeT(float c[12], const float a[12], const float b[12]) {
    float t[12];
#pragma unroll
    for (int r = 0; r < 3; ++r) {
        const float a0 = a[4 * r + 0], a1 = a[4 * r + 1], a2 = a[4 * r + 2], a3 = a[4 * r + 3];
        t[4 * r + 0] = a0 * b[0] + a1 * b[4] + a2 * b[8];
        t[4 * r + 1] = a0 * b[1] + a1 * b[5] + a2 * b[9];
        t[4 * r + 2] = a0 * b[2] + a1 * b[6] + a2 * b[10];
        t[4 * r + 3] = a0 * b[3] + a1 * b[7] + a2 * b[11] + a3;
    }
#pragma unroll
    for (int k = 0; k < 12; ++k) c[k] = t[k];
}

// ---------------------------------------------------------------------------
// LDS layout: cons[0..11]=T0, cons[12..17]=weights (async global->LDS),
//             cons[18..19]=pad, cons[20 + 20*j]=JointC (16B aligned, 80B stride)
// ---------------------------------------------------------------------------
__global__ void __launch_bounds__(256)
fk_fused(const float* __restrict__ thetas,
         const float* __restrict__ target,
         const float* __restrict__ screws,
         const float* __restrict__ weights,
         const float* __restrict__ itw,
         float* __restrict__ out, int B) {
    __shared__ __align__(16) float cons[20 + 6 * 20];
    JointC* joints = (JointC*)(cons + 20);

    {
        const int lt = threadIdx.x;
        if (lt < 6) {
            // CDNA5 async global -> LDS broadcast of the weight vector.
            unsigned lds_off = (unsigned)(unsigned long long)(uintptr_t)(&cons[12 + lt]);
            unsigned long long ga = (unsigned long long)(uintptr_t)(weights + lt);
            asm volatile("global_load_async_to_lds_b32 %0, %1, off"
                         :: "v"(lds_off), "v"(ga) : "memory");
        }
        if (lt == 0) {
            // Uniform inputs -> mostly gfx1250 scalar-float SALU, once per block.
            float tw0[6], M0[12];
#pragma unroll
            for (int k = 0; k < 6; ++k) tw0[k] = itw[k];
            se3_exp6(tw0, M0);
#pragma unroll
            for (int k = 0; k < 12; ++k) cons[k] = M0[k];
#pragma unroll
            for (int j = 0; j < 6; ++j) {
                const float sx = screws[6 * j + 0], sy = screws[6 * j + 1], sz = screws[6 * j + 2];
                const float ux = screws[6 * j + 3], uy = screws[6 * j + 4], uz = screws[6 * j + 5];
                const float k2 = ux * ux + uy * uy + uz * uz;
                const float d  = ux * sx + uy * sy + uz * sz;
                joints[j].u_k     = make_float4(ux, uy, uz, sqrtf(k2));
                joints[j].pd_poxy = make_float4(ux * ux - k2, uy * uy - k2, uz * uz - k2, ux * uy);
                joints[j].po_q1   = make_float4(ux * uz, uy * uz,
                                                uy * sz - uz * sy, uz * sx - ux * sz);
                joints[j].q1z_q2  = make_float4(ux * sy - uy * sx,
                                                ux * d - k2 * sx, uy * d - k2 * sy, uz * d - k2 * sz);
                joints[j].sv      = make_float4(sx, sy, sz, 0.0f);
            }
        }
        asm volatile("s_wait_asynccnt 0" ::: "memory");
        __syncthreads();
    }

    const int i = blockIdx.x * blockDim.x + threadIdx.x;
    if (i >= B) return;

    // Speculative L2 prefetch ~0.75 MB ahead (global_prefetch_b8;
    // translation failures are silently dropped).
    __builtin_prefetch(thetas + (size_t)6 * i + 6 * 32768, 0, 0);
    __builtin_prefetch(target + (size_t)6 * i + 6 * 32768, 0, 0);

    // 6-float records are 8-byte aligned -> three b64 loads each.
    const float2* tp = (const float2*)(thetas + (size_t)6 * i);
    const float2  t01 = tp[0], t23 = tp[1], t45 = tp[2];
    const float   th6[6] = {t01.x, t01.y, t23.x, t23.y, t45.x, t45.y};

    const float2* pp = (const float2*)(target + (size_t)6 * i);
    const float2  p01 = pp[0], p23 = pp[1], p45 = pp[2];
    const float   tgt[6] = {p01.x, p01.y, p23.x, p23.y, p45.x, p45.y};

    const float* T0 = cons;
    const float* W  = cons + 12;

    // ---- forward kinematics: T = exp(s0 th0) ... exp(s5 th5) @ T0 --------
    float T[12], Tj[12];
    joint_exp(&joints[0], th6[0], T);
#pragma unroll
    for (int j = 1; j < 6; ++j) {
        joint_exp(&joints[j], th6[j], Tj);
        composeT(T, T, Tj);
    }
    composeT(T, T, T0);

    // ---- err_T = inv(target_T) @ T ---------------------------------------
    float Tt[12];
    se3_exp6(tgt, Tt);
    const float dx = T[3] - Tt[3], dy = T[7] - Tt[7], dz = T[11] - Tt[11];
    float E[12];
#pragma unroll
    for (int r = 0; r < 3; ++r) {
        // row r of Rt^T is column r of Rt
        const float q0 = Tt[r], q1 = Tt[4 + r], q2 = Tt[8 + r];
        E[4 * r + 0] = q0 * T[0] + q1 * T[4] + q2 * T[8];
        E[4 * r + 1] = q0 * T[1] + q1 * T[5] + q2 * T[9];
        E[4 * r + 2] = q0 * T[2] + q1 * T[6] + q2 * T[10];
        E[4 * r + 3] = q0 * dx + q1 * dy + q2 * dz;
    }

    // ---- se3_log(err_T) ---------------------------------------------------
    const float trc = E[0] + E[5] + E[10];
    float cv = 0.5f * (trc - 1.0f);
    cv = fminf(fmaxf(cv, -1.0f + 1e-7f), 1.0f - 1e-7f);
    const float theta = acosf(cv);
    const bool  small = theta < SE3_EPS;
    const float th = small ? 1.0f : theta;
    float s, c;
    sincosf(th, &s, &c);
    const float coef = small ? (0.5f + theta * theta * (1.0f / 12.0f))
                             : (th / (2.0f * s));
    const float lwx = coef * (E[9] - E[6]);
    const float lwy = coef * (E[2] - E[8]);
    const float lwz = coef * (E[4] - E[1]);
    const float Aa = s / th;
    const float Bb = (1.0f - c) / (th * th);
    const float D  = small ? (1.0f / 12.0f)
                           : ((1.0f - Aa / (2.0f * Bb)) / (th * th));
    const float wn2 = lwx * lwx + lwy * lwy + lwz * lwz;
    const float Vi00 = 1.0f + D * (lwx * lwx - wn2);
    const float Vi01 =  0.5f * lwz + D * lwx * lwy;
    const float Vi02 = -0.5f * lwy + D * lwx * lwz;
    const float Vi10 = -0.5f * lwz + D * lwx * lwy;
    const float Vi11 = 1.0f + D * (lwy * lwy - wn2);
    const float Vi12 =  0.5f * lwx + D * lwy * lwz;
    const float Vi20 =  0.5f * lwy + D * lwx * lwz;
    const float Vi21 = -0.5f * lwx + D * lwy * lwz;
    const float Vi22 = 1.0f + D * (lwz * lwz - wn2);
    const float tx = E[3], ty = E[7], tz = E[11];
    const float lvx = Vi00 * tx + Vi01 * ty + Vi02 * tz;
    const float lvy = Vi10 * tx + Vi11 * ty + Vi12 * tz;
    const float lvz = Vi20 * tx + Vi21 * ty + Vi22 * tz;

    out[i] = fabsf(lvx) * W[0] + fabsf(lvy) * W[1] + fabsf(lvz) * W[2]
           + fabsf(lwx) * W[3] + fabsf(lwy) * W[4] + fabsf(lwz) * W[5];
}

// ---------------------------------------------------------------------------
extern "C" void kernel_launch(void* const* d_in, const int* in_sizes, int n_in,
                              void* d_out, int out_size, void* d_ws, size_t ws_size,
                              hipStream_t stream) {
    const float* thetas  = (const float*)d_in[0];  // (B, 6)
    const float* target  = (const float*)d_in[1];  // (B, 6)
    const float* weights = (const float*)d_in[2];  // (6,)
    const float* screws  = (const float*)d_in[3];  // (6, 6)
    const float* itw     = (const float*)d_in[4];  // (1, 6)
    float*       out     = (float*)d_out;
    (void)d_ws; (void)ws_size; (void)n_in; (void)out_size;

    const int B = in_sizes[0] / 6;
    const int blocks = (B + 255) / 256;
    fk_fused<<<blocks, 256, 0, stream>>>(thetas, target, screws, weights, itw, out, B);
}